// DTWLoss_18854906429643
// MI455X (gfx1250) — compile-verified
//
#include <hip/hip_runtime.h>

typedef __attribute__((ext_vector_type(2))) float v2f;
typedef __attribute__((ext_vector_type(8))) float v8f;

#define INF_F 1.0e30f

// LDS layout (floats):
//  ls1[8192] ls2[8192] lq1[512] lq2[512] Hrow[512] Hcol[512] Hcor[66] Ct[8192]
#define SMEM_FLOATS (8192 + 8192 + 512 + 512 + 512 + 512 + 66 + 8192)

__global__ __launch_bounds__(512) void dtw_batch_kernel(
    const float* __restrict__ s1g, const float* __restrict__ s2g,
    float* __restrict__ res)
{
    extern __shared__ float smem[];
    float* ls1  = smem;            // [512][16] s1 rows of this batch
    float* ls2  = ls1 + 8192;      // [512][16] s2 rows of this batch
    float* lq1  = ls2 + 8192;      // [512] squared norms of s1 rows
    float* lq2  = lq1 + 512;       // [512] squared norms of s2 rows
    float* Hrow = lq2 + 512;       // [512] bottom-row DTW boundary per column j
    float* Hcol = Hrow + 512;      // [512] right-col DTW boundary per row i
    float* Hcor = Hcol + 512;      // [2][33] corner values, parity double-buffered
    float* Ct   = Hcor + 66;       // [16 waves][2 tiles][16][16] cost scratch

    const int tid  = threadIdx.x;
    const int b    = blockIdx.x;
    const int lane = tid & 31;
    const int w    = tid >> 5;     // wave id 0..15
    const int half = lane >> 4;    // 0 or 1 (16-lane half)
    const int l15  = lane & 15;

    // ---- Phase 1: stage s1[b], s2[b] into LDS (float4 vectorized) ----
    const float4* g1 = (const float4*)(s1g + (size_t)b * 8192);
    const float4* g2 = (const float4*)(s2g + (size_t)b * 8192);
    float4* l1v = (float4*)ls1;
    float4* l2v = (float4*)ls2;
    #pragma unroll
    for (int k = 0; k < 4; ++k) {
        int idx = tid + 512 * k;
        l1v[idx] = g1[idx];
        l2v[idx] = g2[idx];
    }
    __syncthreads();

    // ---- norms + boundary init ----
    {
        float a = 0.f, c = 0.f;
        #pragma unroll
        for (int d = 0; d < 16; ++d) {
            float x = ls1[tid * 16 + d]; a += x * x;
            float y = ls2[tid * 16 + d]; c += y * y;
        }
        lq1[tid] = a;  lq2[tid] = c;
        Hrow[tid] = INF_F;  Hcol[tid] = INF_F;
        if (tid < 66) Hcor[tid] = INF_F;
        if (tid == 0) Hcor[0] = 0.f;   // DTW[-1,-1] = 0 start
    }
    __syncthreads();

    // ---- Phase 2: tile-wavefront DTW over 32x32 tiles of 16x16 ----
    for (int td = 0; td < 63; ++td) {
        const int tmin = max(0, td - 31);
        const int tmax = min(td, 31);
        const int nt   = tmax - tmin + 1;
        const int par  = td & 1;

        const int  e0 = 2 * w, e1 = 2 * w + 1;
        const bool valid0 = (e0 < nt), valid1 = (e1 < nt);
        const int  ti0 = tmin + min(e0, nt - 1), tj0 = td - ti0;
        const int  ti1 = tmin + min(e1, nt - 1), tj1 = td - ti1;

        const int  tih = half ? ti1 : ti0;
        const int  tjh = half ? tj1 : tj0;
        const bool validh = half ? valid1 : valid0;
        const int  I0h = tih * 16, J0h = tjh * 16;

        // Phase A: read corner (written same-parity two diagonals ago);
        // barrier protects against this-diagonal overwrite by other tiles.
        const float cornerH = Hcor[par * 33 + tjh];
        __syncthreads();

        // Phase B1: cost tiles via V_WMMA_F32_16X16X4_F32 (K=16 in 4 steps).
        // Norm terms are pre-loaded (vectorized) so their LDS latency overlaps
        // the fragment loads + WMMA pipeline, and the 8 cost stores issue
        // back-to-back without per-element s_wait_dscnt stalls.
        float* ct0 = Ct + w * 512;
        if (valid0) {
            const int I0 = ti0 * 16, J0 = tj0 * 16;
            float  q2 = lq2[J0 + l15];
            float4 qa = *(const float4*)&lq1[I0 + 8 * half];
            float4 qb = *(const float4*)&lq1[I0 + 8 * half + 4];
            v8f acc = {0.f, 0.f, 0.f, 0.f, 0.f, 0.f, 0.f, 0.f};
            #pragma unroll
            for (int q = 0; q < 4; ++q) {
                int k = 4 * q + 2 * half;               // A/B K-split layout
                v2f av = *(const v2f*)&ls1[(I0 + l15) * 16 + k];
                v2f bv = *(const v2f*)&ls2[(J0 + l15) * 16 + k];
                acc = __builtin_amdgcn_wmma_f32_16x16x4_f32(
                          false, av, false, bv, (short)0, acc, false, false);
            }
            float qr[8] = {qa.x, qa.y, qa.z, qa.w, qb.x, qb.y, qb.z, qb.w};
            #pragma unroll
            for (int r = 0; r < 8; ++r) {               // C/D layout: row m, col l15
                ct0[(r + 8 * half) * 16 + l15] = qr[r] + q2 - 2.0f * acc[r];
            }
        }
        if (valid1) {
            const int I0 = ti1 * 16, J0 = tj1 * 16;
            float  q2 = lq2[J0 + l15];
            float4 qa = *(const float4*)&lq1[I0 + 8 * half];
            float4 qb = *(const float4*)&lq1[I0 + 8 * half + 4];
            v8f acc = {0.f, 0.f, 0.f, 0.f, 0.f, 0.f, 0.f, 0.f};
            #pragma unroll
            for (int q = 0; q < 4; ++q) {
                int k = 4 * q + 2 * half;
                v2f av = *(const v2f*)&ls1[(I0 + l15) * 16 + k];
                v2f bv = *(const v2f*)&ls2[(J0 + l15) * 16 + k];
                acc = __builtin_amdgcn_wmma_f32_16x16x4_f32(
                          false, av, false, bv, (short)0, acc, false, false);
            }
            float qr[8] = {qa.x, qa.y, qa.z, qa.w, qb.x, qb.y, qb.z, qb.w};
            #pragma unroll
            for (int r = 0; r < 8; ++r) {
                ct0[256 + (r + 8 * half) * 16 + l15] = qr[r] + q2 - 2.0f * acc[r];
            }
        }
        // wave-local LDS fence: cost tile visible to all lanes of this wave
        asm volatile("s_wait_dscnt 0" ::: "memory");

        // Phase B2: intra-tile systolic DTW, one tile per 16-lane half.
        // Lane = row m; at time t it handles column j = t - m.
        const float* ctH = ct0 + half * 256;
        float leftv    = Hcol[I0h + l15];               // DTW[I0+m, J0-1]
        float leftv_up = __shfl_up(leftv, 1, 16);       // DTW[I0+m-1, J0-1]
        float v_t1 = INF_F, v_t2 = INF_F;
        for (int t = 0; t < 31; ++t) {
            int  j   = t - l15;
            bool act = (j >= 0) & (j < 16);
            int  jc  = min(max(j, 0), 15);
            float upv = __shfl_up(v_t1, 1, 16);         // v(m-1, j)
            float dgv = __shfl_up(v_t2, 1, 16);         // v(m-1, j-1)
            float c   = ctH[l15 * 16 + jc];
            float left = (j == 0) ? leftv : v_t1;
            float up, dg;
            if (l15 == 0) {
                up = Hrow[J0h + jc];                    // top boundary
                dg = (j == 0) ? cornerH : Hrow[J0h + jc - 1];
            } else {
                up = upv;
                dg = (j == 0) ? leftv_up : dgv;
            }
            float v = c + fminf(left, fminf(up, dg));
            if (act) {
                if (validh) {
                    if (l15 == 15) Hrow[J0h + j] = v;           // bottom row out
                    if (j == 15)   Hcol[I0h + l15] = v;         // right col out
                    if (l15 == 15 && j == 15)
                        Hcor[par * 33 + tjh + 1] = v;           // corner out
                }
                v_t2 = v_t1;
                v_t1 = v;
            }
        }
        __syncthreads();
    }

    // tile (31,31) wrote its bottom-right value to Hcor[0][32] at td=62
    if (tid == 0) res[b] = sqrtf(Hcor[0 * 33 + 32]);
}

__global__ void dtw_reduce_kernel(const float* __restrict__ res,
                                  float* __restrict__ out)
{
    if (threadIdx.x == 0) {
        float s = 0.f;
        for (int i = 0; i < 128; ++i) s += res[i];   // fixed order: deterministic
        out[0] = s * (1.0f / 128.0f);
    }
}

extern "C" void kernel_launch(void* const* d_in, const int* in_sizes, int n_in,
                              void* d_out, int out_size, void* d_ws, size_t ws_size,
                              hipStream_t stream) {
    (void)in_sizes; (void)n_in; (void)out_size; (void)ws_size;
    const float* s1 = (const float*)d_in[0];
    const float* s2 = (const float*)d_in[1];
    float* res = (float*)d_ws;    // 128 floats of per-batch sqrt(DTW)
    float* out = (float*)d_out;

    size_t shmem = (size_t)SMEM_FLOATS * sizeof(float);  // ~104 KB of LDS
    dtw_batch_kernel<<<128, 512, shmem, stream>>>(s1, s2, res);
    dtw_reduce_kernel<<<1, 32, 0, stream>>>(res, out);
}